// ADDAAttention_59536836657197
// MI455X (gfx1250) — compile-verified
//
#include <hip/hip_runtime.h>

typedef float v2f __attribute__((ext_vector_type(2)));
typedef float v4f __attribute__((ext_vector_type(4)));
typedef float v8f __attribute__((ext_vector_type(8)));

#define IMG      56
#define HW       (IMG * IMG)          // 3136
#define CHANNELS 512
#define HEADS    8
#define HD       64
#define NPX      16                   // pixels per tile (along W)
#define KCOLS    18                   // union window cols = 16 + 2
#define NSLOT    54                   // 3 * 18 union keys
#define SCALE    0.125f               // 64^-0.5

// LDS strides (bank-conflict-free, 8B-aligned K-pairs)
#define QS_STRIDE 68                  // qs[px][ch]    68 % 64 == 4
#define KS_STRIDE 68                  // ks[slot][ch]  68 % 64 == 4
#define VS_STRIDE 60                  // vs[ch][slot]  60 % 64 == 60 (-4)
#define PS_STRIDE 68                  // ps/ss[px][slot]

__global__ __launch_bounds__(128)
void natten_f32_wmma_kernel(const float* __restrict__ qg,
                            const float* __restrict__ kg,
                            const float* __restrict__ vg,
                            float* __restrict__ outg) {
    __shared__ float qs[NPX * QS_STRIDE];    // [m][c]
    __shared__ float ks[64 * KS_STRIDE];     // [slot][c] rows 54..63 zeroed
    __shared__ float vs[HD * VS_STRIDE];     // [c][slot] slots 54,55 zeroed
    __shared__ float ps[16 * PS_STRIDE];     // [m][slot] probabilities (zero filled)
    __shared__ float ss[16 * PS_STRIDE];     // [m][slot] raw scores

    const int tid  = threadIdx.x;            // 0..127
    const int lane = tid & 31;               // wave32 lane
    const int wv   = tid >> 5;               // 0..3

    int bid = blockIdx.x;
    const int xt   = bid & 3;  bid >>= 2;    // 4 strips of 16 px cover W=56
    const int y    = bid % IMG; bid /= IMG;
    const int head = bid & 7;  bid >>= 3;
    const int b    = bid;
    const int x0   = xt * NPX;

    const size_t chanBase = ((size_t)b * CHANNELS + (size_t)head * HD) * HW;
    const float* qb = qg + chanBase;
    const float* kb = kg + chanBase;
    const float* vb = vg + chanBase;

    // ---------------- stage Q tile (16B vector loads, transposed store) ------
    // idx = c*4 + quarter; each quarter is 4 consecutive pixels
    for (int idx = tid; idx < HD * 4; idx += 128) {
        int c = idx >> 2, qv = idx & 3;
        int x = x0 + 4 * qv;
        v4f val = {0.f, 0.f, 0.f, 0.f};
        if (x < IMG)   // x0 % 16 == 0 and IMG % 4 == 0 -> vector fully in/out
            val = __builtin_nontemporal_load(
                      (const v4f*)(qb + (size_t)c * HW + (size_t)y * IMG + x));
        #pragma unroll
        for (int e = 0; e < 4; ++e)
            qs[(4 * qv + e) * QS_STRIDE + c] = val[e];
    }
    // ---------------- stage K/V union: 3 rows x 18 cols ----------------------
    for (int idx = tid; idx < HD * NSLOT; idx += 128) {
        int c = idx / NSLOT, s = idx % NSLOT;
        int kr = s / KCOLS, kc = s % KCOLS;
        int ky = y - 1 + kr, kx = x0 - 1 + kc;
        bool ok = ((unsigned)ky < (unsigned)IMG) && ((unsigned)kx < (unsigned)IMG);
        size_t g = (size_t)c * HW + (size_t)ky * IMG + kx;
        float kvv = ok ? kb[g] : 0.f;
        float vvv = ok ? vb[g] : 0.f;
        ks[s * KS_STRIDE + c] = kvv;      // [slot][c]
        vs[c * VS_STRIDE + s] = vvv;      // [c][slot]
    }
    // zero padding: ks rows 54..63 (cols 0..63), vs slots 54..55, whole ps
    for (int idx = tid; idx < 10 * HD; idx += 128) {
        int s = 54 + idx / HD, c = idx % HD;
        ks[s * KS_STRIDE + c] = 0.f;
    }
    for (int idx = tid; idx < 2 * HD; idx += 128) {
        int c = idx >> 1, s = 54 + (idx & 1);
        vs[c * VS_STRIDE + s] = 0.f;
    }
    for (int idx = tid; idx < 16 * PS_STRIDE; idx += 128) ps[idx] = 0.f;
    __syncthreads();

    const int half = lane >> 4;      // 0 / 1
    const int nIdx = lane & 15;      // M index for A frags / N index for B,C frags
    const int kOff = half * 2;       // K-dim pairing: lane holds K and K+1; half 1 is at K+2 (pairs K,K+1)

    // ---------------- S = Q * K^T  (each wave owns one 16-key N-tile) --------
    {
        const float* aBase = &qs[nIdx * QS_STRIDE + kOff];             // K = channel
        const float* bBase = &ks[(wv * 16 + nIdx) * KS_STRIDE + kOff]; // N = slot
        v8f acc = {};
        #pragma unroll
        for (int kk = 0; kk < 16; ++kk) {     // 64 channels / 4
            v2f a  = *(const v2f*)(aBase + 4 * kk);
            v2f bf = *(const v2f*)(bBase + 4 * kk);
            acc = __builtin_amdgcn_wmma_f32_16x16x4_f32(false, a, false, bf,
                                                        (short)0, acc, false, false);
        }
        #pragma unroll
        for (int i = 0; i < 8; ++i)
            ss[(i + half * 8) * PS_STRIDE + wv * 16 + nIdx] = acc[i];
    }
    __syncthreads();

    // ---------------- per-pixel masked softmax over 9 neighbors --------------
    if (tid < 16) {
        const int m = tid;
        const int x = x0 + m;
        float sc[9];
        bool  val[9];
        float mx = -3.4e38f;
        #pragma unroll
        for (int kr = 0; kr < 3; ++kr)
            #pragma unroll
            for (int kc = 0; kc < 3; ++kc) {
                int i = kr * 3 + kc;
                int slot = kr * KCOLS + m + kc;
                int ky = y - 1 + kr, kx = x - 1 + kc;
                bool ok = (x < IMG) && ((unsigned)ky < (unsigned)IMG)
                                    && ((unsigned)kx < (unsigned)IMG);
                val[i] = ok;
                float s = ss[m * PS_STRIDE + slot] * SCALE;
                sc[i] = s;
                if (ok && s > mx) mx = s;
            }
        float pr[9];
        float sum = 0.f;
        #pragma unroll
        for (int i = 0; i < 9; ++i) {
            float e = val[i] ? __expf(sc[i] - mx) : 0.f;
            pr[i] = e;
            sum += e;
        }
        float inv = (sum > 0.f) ? 1.f / sum : 0.f;
        #pragma unroll
        for (int kr = 0; kr < 3; ++kr)
            #pragma unroll
            for (int kc = 0; kc < 3; ++kc)
                ps[m * PS_STRIDE + kr * KCOLS + m + kc] = pr[kr * 3 + kc] * inv;
    }
    __syncthreads();

    // ---------------- O = P * V  (each wave owns one 16-channel N-tile) ------
    {
        const float* aBase = &ps[nIdx * PS_STRIDE + kOff];             // K = slot
        const float* bBase = &vs[(wv * 16 + nIdx) * VS_STRIDE + kOff]; // N = channel
        v8f acc = {};
        #pragma unroll
        for (int kk = 0; kk < 14; ++kk) {     // 56 slots / 4
            v2f a  = *(const v2f*)(aBase + 4 * kk);
            v2f bf = *(const v2f*)(bBase + 4 * kk);
            acc = __builtin_amdgcn_wmma_f32_16x16x4_f32(false, a, false, bf,
                                                        (short)0, acc, false, false);
        }
        const size_t outBase = (((size_t)b * IMG + y) * IMG) * CHANNELS
                               + (size_t)head * HD + wv * 16 + nIdx;
        #pragma unroll
        for (int i = 0; i < 8; ++i) {
            int m = i + half * 8;
            int x = x0 + m;
            if (x < IMG)
                __builtin_nontemporal_store(acc[i],
                    outg + outBase + (size_t)x * CHANNELS);
        }
    }
}

extern "C" void kernel_launch(void* const* d_in, const int* in_sizes, int n_in,
                              void* d_out, int out_size, void* d_ws, size_t ws_size,
                              hipStream_t stream) {
    (void)in_sizes; (void)n_in; (void)out_size; (void)d_ws; (void)ws_size;
    const float* q = (const float*)d_in[0];
    const float* k = (const float*)d_in[1];
    const float* v = (const float*)d_in[2];
    float* out = (float*)d_out;

    // grid: B(8) * heads(8) * H(56) * 4 x-strips = 14336 blocks of 128 threads
    const int nBlocks = 8 * HEADS * IMG * 4;
    natten_f32_wmma_kernel<<<nBlocks, 128, 0, stream>>>(q, k, v, out);
}